// HippocampalInhibitoryNetwork_50826642981155
// MI455X (gfx1250) — compile-verified
//
#include <hip/hip_runtime.h>

typedef float v2f __attribute__((ext_vector_type(2)));
typedef float v8f __attribute__((ext_vector_type(8)));

#define N_PYR  40000
#define N_PV   2100
#define N_OLM  1800
#define N_BIST 1200
#define N_SEPT 100

// ---------------------------------------------------------------------------
// Generic fp32 mat-vec via V_WMMA_F32_16X16X4_F32.
//   y[blockIdx.y * Mpad + m] = sum_{k in K-chunk(blockIdx.y)} W[m,k] * x[k]
// One wave handles 16 output rows; 4 waves (64 rows) per workgroup.
// B is the vector broadcast across all 16 N columns, so every column of D
// holds the same dot product; lanes 0 and 16 write rows 0-7 / 8-15.
// Requires K % 4 == 0 (true for all matrices here).
// ---------------------------------------------------------------------------
__global__ __launch_bounds__(128) void wmma_matvec_kernel(
    const float* __restrict__ W, const float* __restrict__ x,
    float* __restrict__ y, int M, int K, int Mpad)
{
  const int lane  = threadIdx.x & 31;
  const int wave  = threadIdx.x >> 5;
  const int rbase = blockIdx.x * 64 + wave * 16;
  if (rbase >= M) return;                 // wave-uniform exit (EXEC stays full)

  const int half = lane >> 4;             // 0: K pair {0,1}, 1: K pair {2,3}
  int row = rbase + (lane & 15);
  row = row < M ? row : (M - 1);          // clamp (duplicates, never written)

  const int S      = gridDim.y;
  const int kquads = K >> 2;
  const int qper   = (kquads + S - 1) / S;
  int k    = blockIdx.y * qper * 4;
  int kend = k + qper * 4;
  kend = kend < K ? kend : K;

  const float* wrow = W + (long long)row * K + 2 * half;
  const float* xh   = x + 2 * half;

  v8f c0 = {}; v8f c1 = {};
  for (; k + 8 <= kend; k += 8) {
    v2f a0 = *(const v2f*)(wrow + k);
    v2f b0 = *(const v2f*)(xh   + k);
    v2f a1 = *(const v2f*)(wrow + k + 4);
    v2f b1 = *(const v2f*)(xh   + k + 4);
    c0 = __builtin_amdgcn_wmma_f32_16x16x4_f32(false, a0, false, b0,
                                               (short)0, c0, false, false);
    c1 = __builtin_amdgcn_wmma_f32_16x16x4_f32(false, a1, false, b1,
                                               (short)0, c1, false, false);
  }
  if (k < kend) {
    v2f a0 = *(const v2f*)(wrow + k);
    v2f b0 = *(const v2f*)(xh   + k);
    c0 = __builtin_amdgcn_wmma_f32_16x16x4_f32(false, a0, false, b0,
                                               (short)0, c0, false, false);
  }
  v8f c = c0 + c1;

  if ((lane & 15) == 0) {                 // lanes 0 and 16: column N==0
    const int mb = rbase + (half << 3);   // rows 0-7 or 8-15 of this block
    float* dst = y + (long long)blockIdx.y * Mpad + mb;
#pragma unroll
    for (int j = 0; j < 8; ++j)
      if (mb + j < M) dst[j] = c[j];
  }
}

// out[i] = sum_s part[s*Mpad + i]
__global__ void combine_kernel(const float* __restrict__ part,
                               float* __restrict__ out, int M, int S, int Mpad)
{
  int i = blockIdx.x * blockDim.x + threadIdx.x;
  if (i >= M) return;
  float s = 0.f;
  for (int j = 0; j < S; ++j) s += part[(long long)j * Mpad + i];
  out[i] = s;
}

__global__ void i2f_kernel(const int* __restrict__ in, float* __restrict__ out, int n)
{
  int i = blockIdx.x * blockDim.x + threadIdx.x;
  if (i < n) out[i] = (float)in[i];
}

// scal[0] = mean(external_exc); scal[1] = mean(pv_exc_from_pyr)   (N_PV each)
__global__ void means_kernel(const float* __restrict__ ext,
                             const float* __restrict__ pvraw,
                             float* __restrict__ scal)
{
  __shared__ float sh[256];
  float s0 = 0.f, s1 = 0.f;
  for (int i = threadIdx.x; i < N_PV; i += 256) { s0 += ext[i]; s1 += pvraw[i]; }
  sh[threadIdx.x] = s0; __syncthreads();
  for (int o = 128; o > 0; o >>= 1) {
    if (threadIdx.x < o) sh[threadIdx.x] += sh[threadIdx.x + o];
    __syncthreads();
  }
  if (threadIdx.x == 0) scal[0] = sh[0] * (1.0f / N_PV);
  __syncthreads();
  sh[threadIdx.x] = s1; __syncthreads();
  for (int o = 128; o > 0; o >>= 1) {
    if (threadIdx.x < o) sh[threadIdx.x] += sh[threadIdx.x + o];
    __syncthreads();
  }
  if (threadIdx.x == 0) scal[1] = sh[0] * (1.0f / N_PV);
}

// Fused conductance assembly + LIF spike for all three populations.
__global__ void lif_kernel(const float* __restrict__ pv_raw,
                           const float* __restrict__ olm_raw,
                           const float* __restrict__ bist_raw,
                           const float* __restrict__ pv_lat,
                           const float* __restrict__ olm2pv,
                           const float* __restrict__ gap,
                           const float* __restrict__ olm_ginh,
                           const float* __restrict__ scal,
                           const float* __restrict__ pv_v,
                           const float* __restrict__ olm_v,
                           const float* __restrict__ bist_v,
                           const float* __restrict__ pv_ad,
                           const float* __restrict__ olm_ad,
                           const float* __restrict__ bist_ad,
                           float* __restrict__ pv_spk,
                           float* __restrict__ olm_spk,
                           float* __restrict__ bist_spk)
{
  const int i = blockIdx.x * blockDim.x + threadIdx.x;
  const float ext_mean = scal[0];
  const float drive    = scal[1];
  const float gain = (drive > 0.01f) ? fminf(1.0f, drive * 10.0f) : 0.0f;

  if (i < N_PV) {
    float g = pv_raw[i] + ext_mean * 0.3f
            - pv_lat[i] * 0.5f - olm2pv[i] * 0.3f
            + gap[i] * 0.05f * gain;
    g = fmaxf(g, 0.0f);
    const float v  = pv_v[i];
    const float vn = v + (1.0f / 7.0f) * (-v + g * (3.0f - v));
    pv_spk[i] = (vn >= 2.0f + pv_ad[i]) ? 1.0f : 0.0f;
  }
  if (i < N_OLM) {
    float g = fmaxf(olm_raw[i] + ext_mean * 0.1f, 0.0f);
    const float gi = olm_ginh[i];
    const float v  = olm_v[i];
    const float vn = v + (1.0f / 25.0f) * (-v + g * (3.0f - v) + gi * (-0.5f - v));
    olm_spk[i] = (vn >= 1.1f + olm_ad[i]) ? 1.0f : 0.0f;
  }
  if (i < N_BIST) {
    float g = fmaxf(bist_raw[i] + ext_mean * 0.2f, 0.0f);
    const float v  = bist_v[i];
    const float vn = v + (1.0f / 12.0f) * (-v + g * (3.0f - v));
    bist_spk[i] = (vn >= 0.9f + bist_ad[i]) ? 1.0f : 0.0f;
  }
}

__global__ void add_kernel(float* __restrict__ a, const float* __restrict__ b, int n)
{
  int i = blockIdx.x * blockDim.x + threadIdx.x;
  if (i < n) a[i] += b[i];
}

static inline int round64(int m) { return ((m + 63) / 64) * 64; }

extern "C" void kernel_launch(void* const* d_in, const int* in_sizes, int n_in,
                              void* d_out, int out_size, void* d_ws, size_t ws_size,
                              hipStream_t stream)
{
  (void)in_sizes; (void)n_in; (void)out_size; (void)ws_size;

  // ---- inputs (setup_inputs order) ----
  const int*   pyr_spikes      = (const int*)  d_in[0];
  const float* septal_gaba     = (const float*)d_in[1];
  const float* external_exc    = (const float*)d_in[2];
  const float* prev_pv_spikes  = (const float*)d_in[3];
  const float* prev_olm_spikes = (const float*)d_in[4];
  const float* prev_pv_v_mem   = (const float*)d_in[5];
  const float* pv_v            = (const float*)d_in[6];
  const float* olm_v           = (const float*)d_in[7];
  const float* bist_v          = (const float*)d_in[8];
  const float* pv_adapt        = (const float*)d_in[9];
  const float* olm_adapt       = (const float*)d_in[10];
  const float* bist_adapt      = (const float*)d_in[11];
  const float* pyr_to_pv       = (const float*)d_in[12];
  const float* pyr_to_olm      = (const float*)d_in[13];
  const float* pyr_to_bist     = (const float*)d_in[14];
  const float* pv_to_pyr       = (const float*)d_in[15];
  const float* pv_to_pyr_gb    = (const float*)d_in[16];
  const float* olm_to_pyr      = (const float*)d_in[17];
  const float* bist_to_pyr     = (const float*)d_in[18];
  const float* pv_to_pv        = (const float*)d_in[19];
  const float* olm_to_pv       = (const float*)d_in[20];
  const float* pv_gap          = (const float*)d_in[21];
  const float* septal_to_olm   = (const float*)d_in[22];

  // ---- output layout (flat, return order) ----
  float* out         = (float*)d_out;
  float* perisomatic = out;                // 40000
  float* peri_gaba_b = out + 1 * N_PYR;    // 40000
  float* dendritic   = out + 2 * N_PYR;    // 40000 (first holds bist part)
  float* olm_dend    = out + 3 * N_PYR;    // 40000
  float* pv_spk      = out + 4 * N_PYR;            // 2100
  float* olm_spk     = pv_spk + N_PV;              // 1800
  float* bist_spk    = olm_spk + N_OLM;            // 1200

  // ---- workspace layout (floats) ----
  const int S_BIG = 32, S_SMALL = 4;
  const int MP_PV = round64(N_PV), MP_OLM = round64(N_OLM), MP_BI = round64(N_BIST);
  float* ws = (float*)d_ws;
  size_t o = 0;
  float* pyr_f     = ws + o; o += N_PYR;
  float* part_pv   = ws + o; o += (size_t)S_BIG * MP_PV;
  float* part_olm  = ws + o; o += (size_t)S_BIG * MP_OLM;
  float* part_bi   = ws + o; o += (size_t)S_BIG * MP_BI;
  float* part_lat  = ws + o; o += (size_t)S_SMALL * MP_PV;
  float* part_o2p  = ws + o; o += (size_t)S_SMALL * MP_PV;
  float* part_gap  = ws + o; o += (size_t)S_SMALL * MP_PV;
  float* pv_raw    = ws + o; o += N_PV;
  float* olm_raw   = ws + o; o += N_OLM;
  float* bist_raw  = ws + o; o += N_BIST;
  float* pv_lat    = ws + o; o += N_PV;
  float* olm2pv    = ws + o; o += N_PV;
  float* gap_cur   = ws + o; o += N_PV;
  float* olm_ginh  = ws + o; o += N_OLM;
  float* scal      = ws + o; o += 2;

  // ---- 1. pyr spikes int32 -> float ----
  i2f_kernel<<<(N_PYR + 255) / 256, 256, 0, stream>>>(pyr_spikes, pyr_f, N_PYR);

  // ---- 2. interneuron-side mat-vecs (K-split for occupancy) ----
  dim3 blk(128);
  wmma_matvec_kernel<<<dim3((N_PV  + 63) / 64, S_BIG), blk, 0, stream>>>(
      pyr_to_pv,   pyr_f, part_pv,  N_PV,  N_PYR, MP_PV);
  wmma_matvec_kernel<<<dim3((N_OLM + 63) / 64, S_BIG), blk, 0, stream>>>(
      pyr_to_olm,  pyr_f, part_olm, N_OLM, N_PYR, MP_OLM);
  wmma_matvec_kernel<<<dim3((N_BIST + 63) / 64, S_BIG), blk, 0, stream>>>(
      pyr_to_bist, pyr_f, part_bi,  N_BIST, N_PYR, MP_BI);
  wmma_matvec_kernel<<<dim3((N_PV + 63) / 64, S_SMALL), blk, 0, stream>>>(
      pv_to_pv,  prev_pv_spikes,  part_lat, N_PV, N_PV,  MP_PV);
  wmma_matvec_kernel<<<dim3((N_PV + 63) / 64, S_SMALL), blk, 0, stream>>>(
      olm_to_pv, prev_olm_spikes, part_o2p, N_PV, N_OLM, MP_PV);
  wmma_matvec_kernel<<<dim3((N_PV + 63) / 64, S_SMALL), blk, 0, stream>>>(
      pv_gap,    prev_pv_v_mem,   part_gap, N_PV, N_PV,  MP_PV);
  wmma_matvec_kernel<<<dim3((N_OLM + 63) / 64, 1), blk, 0, stream>>>(
      septal_to_olm, septal_gaba, olm_ginh, N_OLM, N_SEPT, N_OLM);

  // ---- 3. combine K-split partials ----
  combine_kernel<<<(N_PV   + 255) / 256, 256, 0, stream>>>(part_pv,  pv_raw,   N_PV,  S_BIG,   MP_PV);
  combine_kernel<<<(N_OLM  + 255) / 256, 256, 0, stream>>>(part_olm, olm_raw,  N_OLM, S_BIG,   MP_OLM);
  combine_kernel<<<(N_BIST + 255) / 256, 256, 0, stream>>>(part_bi,  bist_raw, N_BIST, S_BIG,  MP_BI);
  combine_kernel<<<(N_PV   + 255) / 256, 256, 0, stream>>>(part_lat, pv_lat,   N_PV,  S_SMALL, MP_PV);
  combine_kernel<<<(N_PV   + 255) / 256, 256, 0, stream>>>(part_o2p, olm2pv,   N_PV,  S_SMALL, MP_PV);
  combine_kernel<<<(N_PV   + 255) / 256, 256, 0, stream>>>(part_gap, gap_cur,  N_PV,  S_SMALL, MP_PV);

  // ---- 4. scalar means (ext_mean, drive) ----
  means_kernel<<<1, 256, 0, stream>>>(external_exc, pv_raw, scal);

  // ---- 5. LIF spikes -> d_out tail ----
  lif_kernel<<<(N_PV + 255) / 256, 256, 0, stream>>>(
      pv_raw, olm_raw, bist_raw, pv_lat, olm2pv, gap_cur, olm_ginh, scal,
      pv_v, olm_v, bist_v, pv_adapt, olm_adapt, bist_adapt,
      pv_spk, olm_spk, bist_spk);

  // ---- 6. pyr-side mat-vecs (2500 row-blocks each; no K-split needed) ----
  wmma_matvec_kernel<<<dim3((N_PYR + 63) / 64, 1), blk, 0, stream>>>(
      pv_to_pyr,    pv_spk,   perisomatic, N_PYR, N_PV,   N_PYR);
  wmma_matvec_kernel<<<dim3((N_PYR + 63) / 64, 1), blk, 0, stream>>>(
      pv_to_pyr_gb, pv_spk,   peri_gaba_b, N_PYR, N_PV,   N_PYR);
  wmma_matvec_kernel<<<dim3((N_PYR + 63) / 64, 1), blk, 0, stream>>>(
      bist_to_pyr,  bist_spk, dendritic,   N_PYR, N_BIST, N_PYR);
  wmma_matvec_kernel<<<dim3((N_PYR + 63) / 64, 1), blk, 0, stream>>>(
      olm_to_pyr,   olm_spk,  olm_dend,    N_PYR, N_OLM,  N_PYR);

  // ---- 7. dendritic = bist part + olm_dendritic ----
  add_kernel<<<(N_PYR + 255) / 256, 256, 0, stream>>>(dendritic, olm_dend, N_PYR);
}